// WaveletTransformLayer_82849919140027
// MI455X (gfx1250) — compile-verified
//
#include <hip/hip_runtime.h>
#include <hip/hip_bf16.h>
#include <math.h>

// ---------------------------------------------------------------------------
// WaveletTransformLayer for MI455X (gfx1250, wave32)
//  Kernel 1: per-row 7-level DWT + soft threshold + IDWT, fully in LDS.
//  Kernel 2: recon @ W1^T + b1 -> sigmoid -> @ W2^T + b2 using
//            V_WMMA_F32_16X16X4_F32, with TDM (tensor_load_to_lds)
//            double-buffered A-chunk staging: the D# descriptor's
//            tensor_dim0=kmax / tile_dim0=116 does K-tail zero-padding and
//            the LDS pitch in hardware, and s_wait_tensorcnt(1) overlaps the
//            next chunk's DMA with the current 28-WMMA burst.
// ---------------------------------------------------------------------------

typedef __attribute__((ext_vector_type(2))) float v2f;
typedef __attribute__((ext_vector_type(8))) float v8f;
typedef unsigned int uint32x4 __attribute__((ext_vector_type(4)));
typedef int int32x8 __attribute__((ext_vector_type(8)));
typedef int int32x4 __attribute__((ext_vector_type(4)));

__constant__ float c_lo[8] = {
    -0.010597401784997278f,  0.032883011666982945f,  0.030841381835986965f,
    -0.18703481171888114f,  -0.02798376941698385f,   0.6308807679295904f,
     0.7148465705525415f,    0.23037781330885523f};
__constant__ float c_hi[8] = {
    -0.23037781330885523f,   0.7148465705525415f,   -0.6308807679295904f,
    -0.02798376941698385f,   0.18703481171888114f,   0.030841381835986965f,
    -0.032883011666982945f, -0.010597401784997278f};

__device__ __forceinline__ float softt(float c, float t) {
    float a = fabsf(c) - t;
    a = a > 0.0f ? a : 0.0f;
    return copysignf(a, c);
}

// ---------------------------------------------------------------------------
// Kernel 1: one workgroup per row. N <= 1152, LEVEL = 7.
// ---------------------------------------------------------------------------
__global__ __launch_bounds__(256) void wavelet_kernel(
    const float* __restrict__ x, const float* __restrict__ thr,
    float* __restrict__ recon, int N) {
    __shared__ float abuf[1152];   // current approximation (max N)
    __shared__ float ybuf[1152];   // scratch / idwt output
    __shared__ float dbuf[1200];   // all detail coeffs concatenated (1146)

    const int row = blockIdx.x;
    const float t = fmaxf(thr[0], 0.01f);
    const float* xr = x + (size_t)row * N;

    const int Nq = N >> 2;
    for (int i = threadIdx.x; i < Nq; i += blockDim.x)
        *(float4*)&abuf[i * 4] = ((const float4*)xr)[i];
    for (int i = (Nq << 2) + threadIdx.x; i < N; i += blockDim.x)
        abuf[i] = xr[i];
    __syncthreads();

    int lens[8];
    int doff[7];
    lens[0] = N;
    int off = 0;
    for (int l = 0; l < 7; ++l) {
        lens[l + 1] = (lens[l] + 5) / 2 + 1;
        doff[l] = off;
        off += lens[l + 1];
    }

    // ---- forward DWT: a[i] = sum_s lo[s]*x[2i+1-s] (zero-padded) ----
    for (int l = 0; l < 7; ++l) {
        const int n = lens[l];
        const int m = lens[l + 1];
        float* d = dbuf + doff[l];
        const int iTop = (n - 2) / 2 + 1;   // interior: 3 <= i < iTop

        for (int i = threadIdx.x; i < 3 && i < m; i += blockDim.x) {
            float sa = 0.0f, sd = 0.0f;
            const int base = 2 * i + 1;
#pragma unroll
            for (int s = 0; s < 8; ++s) {
                const int ix = base - s;
                if (ix >= 0) {
                    const float v = abuf[ix];
                    sa = fmaf(c_lo[s], v, sa);
                    sd = fmaf(c_hi[s], v, sd);
                }
            }
            ybuf[i] = sa;
            d[i] = softt(sd, t);
        }
        for (int i = 3 + threadIdx.x; i < iTop && i < m; i += blockDim.x) {
            const float* xb = &abuf[2 * i + 1];
            float sa = 0.0f, sd = 0.0f;
#pragma unroll
            for (int s = 0; s < 8; ++s) {
                const float v = xb[-s];
                sa = fmaf(c_lo[s], v, sa);
                sd = fmaf(c_hi[s], v, sd);
            }
            ybuf[i] = sa;
            d[i] = softt(sd, t);
        }
        for (int i = (iTop > 3 ? iTop : 3) + threadIdx.x; i < m; i += blockDim.x) {
            float sa = 0.0f, sd = 0.0f;
            const int base = 2 * i + 1;
#pragma unroll
            for (int s = 0; s < 8; ++s) {
                const int ix = base - s;
                if (ix < n) {
                    const float v = abuf[ix];
                    sa = fmaf(c_lo[s], v, sa);
                    sd = fmaf(c_hi[s], v, sd);
                }
            }
            ybuf[i] = sa;
            d[i] = softt(sd, t);
        }
        __syncthreads();
        for (int i = threadIdx.x; i < m; i += blockDim.x) abuf[i] = ybuf[i];
        __syncthreads();
    }

    for (int i = threadIdx.x; i < lens[7]; i += blockDim.x)
        abuf[i] = softt(abuf[i], t);
    __syncthreads();

    // ---- inverse DWT: guard-free (all 4 odd-tap indices in [0, nd-1]) ----
    for (int l = 6; l >= 0; --l) {
        const int nd = lens[l + 1];
        const float* d = dbuf + doff[l];
        const int ny = 2 * nd - 6;
        for (int j = threadIdx.x; j < ny; j += blockDim.x) {
            const int h = j >> 1;
            const int t0 = (j & 1) ? 0 : 1;
            float s = 0.0f;
#pragma unroll
            for (int q = 0; q < 4; ++q) {
                const int tt = 2 * q + t0;
                s = fmaf(c_lo[tt], abuf[h + q], s);
                s = fmaf(c_hi[tt], d[h + q], s);
            }
            ybuf[j] = s;
        }
        __syncthreads();
        for (int j = threadIdx.x; j < ny; j += blockDim.x) abuf[j] = ybuf[j];
        __syncthreads();
    }

    float* rr = recon + (size_t)row * N;
    for (int i = threadIdx.x; i < Nq; i += blockDim.x)
        ((float4*)rr)[i] = *(const float4*)&abuf[i * 4];
    for (int i = (Nq << 2) + threadIdx.x; i < N; i += blockDim.x)
        rr[i] = abuf[i];
}

// ---------------------------------------------------------------------------
// Kernel 2: 1 block = 16 rows, 7 waves (one 16-wide N tile each, 112 >= H).
// ---------------------------------------------------------------------------
#define KC 112
#define KQ (KC / 4)
#define APITCH 116    // 16B-aligned rows; lanes 0..15 hit 16 distinct banks
#define ABUF (16 * APITCH)
#define HPITCH 104

#if defined(__has_builtin)
#if __has_builtin(__builtin_amdgcn_tensor_load_to_lds) && \
    __has_builtin(__builtin_amdgcn_s_wait_tensorcnt)
#define USE_TDM 1
#endif
#endif

#ifdef USE_TDM
// Issue one TDM 2D tile load: 16 rows x tile_dim0=116 elems (f32) from
// gsrc (row stride = strideElems) into LDS at lds_byte_addr. Columns
// >= kmax are outside tensor_dim0 -> TDM writes zeros (K-tail padding and
// LDS pitch fill in hardware). D# packing per CDNA5 ISA.
// Toolchain here is the 6-arg clang-23 form:
//   (uint32x4 g0, int32x8 g1, int32x4 g2, int32x4 g3, int32x8, i32 cpol)
__device__ __forceinline__ void tdm_load_chunk(const float* gsrc,
                                               unsigned int lds_byte_addr,
                                               int kmax, int strideElems) {
    const unsigned long long ga = (unsigned long long)(uintptr_t)gsrc;
    uint32x4 g0;
    g0[0] = 1u;                                    // count=1, user mode, no gather
    g0[1] = lds_byte_addr;                         // D#.lds_addr
    g0[2] = (unsigned int)ga;                      // global_addr[31:0]
    g0[3] = (unsigned int)((ga >> 32) & 0x01FFFFFFu) | (2u << 30);  // [56:32]|type=2
    int32x8 g1;
    g1[0] = (2 << 16);                             // data_size = 4 bytes
    g1[1] = (int)(((unsigned int)kmax & 0xFFFFu) << 16);  // tensor_dim0[15:0]
    g1[2] = (16 << 16);                            // tensor_dim0 hi | tensor_dim1=16
    g1[3] = (APITCH << 16);                        // tensor_dim1 hi | tile_dim0=116
    g1[4] = 16;                                    // tile_dim1=16, tile_dim2=0
    g1[5] = strideElems;                           // tensor_dim0_stride[31:0]
    g1[6] = 0;                                     // stride hi | dim1_stride lo
    g1[7] = 0;
    const int32x4 z4 = {0, 0, 0, 0};               // 2D tensor: groups 2/3 unused
    const int32x8 z8 = {0, 0, 0, 0, 0, 0, 0, 0};
    __builtin_amdgcn_tensor_load_to_lds(g0, g1, z4, z4, z8, 0);
}
#endif

__device__ __forceinline__ v8f wmma_step(const float* ak, const float* wk,
                                         int kk, v8f c) {
    v2f a, b;
    a.x = ak[kk];
    a.y = ak[kk + 1];
    b = *(const v2f*)(wk + kk);
    return __builtin_amdgcn_wmma_f32_16x16x4_f32(
        false, a, false, b, (short)0, c, false, false);
}

__global__ __launch_bounds__(224) void mlp_kernel(
    const float* __restrict__ recon, const float* __restrict__ W1,
    const float* __restrict__ b1, const float* __restrict__ W2,
    const float* __restrict__ b2, float* __restrict__ reg,
    int N, int H) {
    __shared__ float As[2][ABUF];         // double-buffered 16 x KC A chunks
    __shared__ float hbuf[16 * HPITCH];   // sigmoid(h) tile, 16 x H

    const int tid  = threadIdx.x;
    const int lane = tid & 31;
    const int wave = tid >> 5;
    const int row0 = blockIdx.x * 16;

    const int half = lane >> 4;           // 0: K={0,1}, 1: K={2,3} (A frag)
    const int nloc = lane & 15;
    const int n_global = wave * 16 + nloc;
    const int koff = half * 2;
    const int mA   = nloc;                // A-frag row for this lane

    // Dead lanes (n_global >= H) read a clamped valid W1 row: WMMA column n
    // only feeds output column n, which is never stored for dead lanes.
    const int nW = (n_global < H) ? n_global : (H - 1);
    const float* w1base = W1 + (size_t)nW * N + koff;
    const float* reconRow0 = recon + (size_t)row0 * N;

    v8f c;
#pragma unroll
    for (int r = 0; r < 8; ++r) c[r] = 0.0f;

    const int nchunks = (N + KC - 1) / KC;   // 10 for N=1116

#ifdef USE_TDM
    const unsigned int ldsA0 = (unsigned int)(uintptr_t)&As[0][0];
    const unsigned int ldsA1 = (unsigned int)(uintptr_t)&As[1][0];
    if (wave == 0) {   // TDM issue is per-wave; EXEC ignored by tensor ops
        const int km0 = (N < KC) ? N : KC;
        tdm_load_chunk(reconRow0, ldsA0, km0, N);
    }
#endif

    for (int ch = 0; ch < nchunks; ++ch) {
        const int kbase = ch * KC;
#ifdef USE_TDM
        if (wave == 0) {
            if (ch + 1 < nchunks) {
                const int kb2 = kbase + KC;
                const int km2 = (N - kb2 < KC) ? (N - kb2) : KC;
                tdm_load_chunk(reconRow0 + kb2, ((ch + 1) & 1) ? ldsA1 : ldsA0,
                               km2, N);
                __builtin_amdgcn_s_wait_tensorcnt(1);  // chunk ch landed
            } else {
                __builtin_amdgcn_s_wait_tensorcnt(0);
            }
        }
        __syncthreads();                      // buf[ch&1] visible to all waves
#else
        __syncthreads();
        // fallback: synchronous float4 staging with zero-pad
        for (int idx = tid; idx < 16 * KQ; idx += blockDim.x) {
            const int m = idx / KQ;
            const int kq = idx % KQ;
            float4 v = make_float4(0.f, 0.f, 0.f, 0.f);
            if (kbase + kq * 4 < N)   // K tail is a multiple of 4
                v = *(const float4*)&reconRow0[(size_t)m * N + kbase + kq * 4];
            *(float4*)&As[ch & 1][m * APITCH + kq * 4] = v;
        }
        __syncthreads();
#endif
        const float* akb = &As[ch & 1][0] + mA * APITCH + koff;
        const float* wk = w1base + kbase;
        if (kbase + KC <= N) {
#pragma unroll
            for (int kk = 0; kk < KC; kk += 4)   // 28 back-to-back WMMAs
                c = wmma_step(akb, wk, kk, c);
        } else {
            const int kmax = N - kbase;          // multiple of 4
#pragma unroll 4
            for (int kk = 0; kk + 3 < kmax; kk += 4)
                c = wmma_step(akb, wk, kk, c);
        }
        __syncthreads();   // done reading buf[ch&1] before TDM overwrites it
    }

    // bias + sigmoid, drop h tile to LDS
    if (n_global < H) {
        const float bb = b1[n_global];
#pragma unroll
        for (int r = 0; r < 8; ++r) {
            const int m = r + 8 * half;   // C/D layout: VGPR r, lane half
            const float hv = 1.0f / (1.0f + expf(-(c[r] + bb)));
            hbuf[m * HPITCH + n_global] = hv;
        }
    }
    __syncthreads();

    // reg[m] = b2 + sum_j h[m][j] * W2[j]   (H = 100, trivial)
    if (tid < 16) {
        float s = b2[0];
        for (int j = 0; j < H; ++j) s += hbuf[tid * HPITCH + j] * W2[j];
        reg[row0 + tid] = s;
    }
}

extern "C" void kernel_launch(void* const* d_in, const int* in_sizes, int n_in,
                              void* d_out, int out_size, void* d_ws, size_t ws_size,
                              hipStream_t stream) {
    const float* x   = (const float*)d_in[0];
    const float* thr = (const float*)d_in[1];
    const float* W1  = (const float*)d_in[2];
    const float* b1  = (const float*)d_in[3];
    const float* W2  = (const float*)d_in[4];
    const float* b2  = (const float*)d_in[5];

    const int H = in_sizes[3];            // 100
    const int N = in_sizes[2] / H;        // 1116
    const int B = in_sizes[0] / N;        // 32768

    float* recon = (float*)d_out;                      // B*N floats
    float* reg   = (float*)d_out + (size_t)B * N;      // B floats

    wavelet_kernel<<<B, 256, 0, stream>>>(x, thr, recon, N);
    mlp_kernel<<<B / 16, 224, 0, stream>>>(recon, W1, b1, W2, b2, reg, N, H);
}